// HungarianMatcher_9706626089176
// MI455X (gfx1250) — compile-verified
//
#include <hip/hip_runtime.h>
#include <hip/hip_bf16.h>

// HungarianMatcher cost matrix, fused single pass.
//   C[i,j] = -softmax(logits)[i, id[j]] + 5*L1(cxcywh) - 2*GIoU(xyxy)
// Class term is a one-hot matmul executed on v_wmma_f32_16x16x4_f32 (exact f32),
// seeded with the VALU-computed 5*L1 - 2*GIoU accumulator.

typedef __attribute__((ext_vector_type(2))) float v2f;
typedef __attribute__((ext_vector_type(4))) float v4f;
typedef __attribute__((ext_vector_type(8))) float v8f;

#define NBATCH   16
#define NQ       1000
#define NC       92
#define NT       1600
#define NROWS    (NBATCH * NQ)        // 16000 query rows
#define KSLICES  (NC / 4)             // 23 K-slices of 4 for 16x16x4 WMMA
#define NSTRIPS  (NROWS / 16)         // 1000 row strips
#define TILES_PER_CHUNK 10            // column tiles (of 16) per wave
#define NCHUNKS  (NT / 16 / TILES_PER_CHUNK)   // 10
#define WAVES_PER_BLOCK 8
#define NTASKS   (NSTRIPS * NCHUNKS)  // 10000 waves

__global__ __launch_bounds__(WAVES_PER_BLOCK * 32)
void matcher_cost_kernel(const float* __restrict__ logits,   // (16000, 92)
                         const float* __restrict__ pboxes,   // (16000, 4) cxcywh
                         const int*   __restrict__ tids,     // (1600,)
                         const float* __restrict__ tboxes,   // (1600, 4) cxcywh
                         float*       __restrict__ out)      // (16000, 1600)
{
    const int lane = threadIdx.x & 31;
    const int wave = threadIdx.x >> 5;
    const int task = blockIdx.x * WAVES_PER_BLOCK + wave;
    if (task >= NTASKS) return;               // wave-uniform; EXEC stays all-ones

    const int strip = task / NCHUNKS;         // 0..999
    const int chunk = task % NCHUNKS;         // 0..9
    const int row_base = strip * 16;
    const int half  = lane >> 4;              // 0: M 0..7 / K 0..1 ; 1: M 8..15 / K 2..3
    const int nlane = lane & 15;              // A-row within strip == target column in tile
    const int khalf = half * 2;

    // ---- A tile: logits -> in-register softmax -> negated probs, WMMA layout ----
    // A[m][k] lives at lane = m + 16*(k>=2), vgpr = k&1  (32-bit A 16x4 layout)
    v2f a[KSLICES];
    {
        const float* lrow = logits + (size_t)(row_base + nlane) * NC;
        float mx = -3.402823466e38f;
        #pragma unroll
        for (int s = 0; s < KSLICES; ++s) {
            a[s] = *(const v2f*)(lrow + 4 * s + khalf);   // 8B-aligned (row stride 368B)
            mx = fmaxf(mx, fmaxf(a[s].x, a[s].y));
        }
        mx = fmaxf(mx, __shfl_xor(mx, 16, 32));           // combine lane m with m+16
        float sum = 0.0f;
        #pragma unroll
        for (int s = 0; s < KSLICES; ++s) {
            a[s].x = __expf(a[s].x - mx);
            a[s].y = __expf(a[s].y - mx);
            sum += a[s].x + a[s].y;
        }
        sum += __shfl_xor(sum, 16, 32);
        const float nscale = -1.0f / sum;                 // A = -prob
        #pragma unroll
        for (int s = 0; s < KSLICES; ++s) { a[s].x *= nscale; a[s].y *= nscale; }
    }

    // ---- query boxes for this lane's 8 C-rows (M = r + 8*half) ----
    float qcx[8], qcy[8], qw[8], qh[8];
    float qx0[8], qy0[8], qx1[8], qy1[8], qarea[8];
    {
        const float* qb = pboxes + (size_t)(row_base + half * 8) * 4;
        #pragma unroll
        for (int r = 0; r < 8; ++r) {
            v4f b = *(const v4f*)(qb + r * 4);
            qcx[r] = b.x; qcy[r] = b.y; qw[r] = b.z; qh[r] = b.w;
            qx0[r] = b.x - 0.5f * b.z;  qy0[r] = b.y - 0.5f * b.w;
            qx1[r] = b.x + 0.5f * b.z;  qy1[r] = b.y + 0.5f * b.w;
            qarea[r] = b.z * b.w;
        }
    }

    const int col_base0 = chunk * TILES_PER_CHUNK * 16;

    #pragma unroll 1
    for (int t = 0; t < TILES_PER_CHUNK; ++t) {
        const int col = col_base0 + t * 16 + nlane;       // this lane's target column
        const v4f tb  = *(const v4f*)(tboxes + (size_t)col * 4);
        const int tid = tids[col];
        const float tx0 = tb.x - 0.5f * tb.z, ty0 = tb.y - 0.5f * tb.w;
        const float tx1 = tb.x + 0.5f * tb.z, ty1 = tb.y + 0.5f * tb.w;
        const float tarea = tb.z * tb.w;

        // C seed: 5*L1 - 2*GIoU for the 8 (M, N=col) pairs this lane owns
        v8f acc;
        #pragma unroll
        for (int r = 0; r < 8; ++r) {
            const float l1 = fabsf(qcx[r] - tb.x) + fabsf(qcy[r] - tb.y)
                           + fabsf(qw[r]  - tb.z) + fabsf(qh[r]  - tb.w);
            const float ix0 = fmaxf(qx0[r], tx0), iy0 = fmaxf(qy0[r], ty0);
            const float ix1 = fminf(qx1[r], tx1), iy1 = fminf(qy1[r], ty1);
            const float iw = fmaxf(ix1 - ix0, 0.0f), ih = fmaxf(iy1 - iy0, 0.0f);
            const float inter = iw * ih;
            const float uni   = qarea[r] + tarea - inter;
            const float iou   = inter / uni;
            const float ex0 = fminf(qx0[r], tx0), ey0 = fminf(qy0[r], ty0);
            const float ex1 = fmaxf(qx1[r], tx1), ey1 = fmaxf(qy1[r], ty1);
            const float ew = fmaxf(ex1 - ex0, 0.0f), eh = fmaxf(ey1 - ey0, 0.0f);
            const float earea = ew * eh;
            const float giou  = iou - (earea - uni) / earea;
            acc[r] = 5.0f * l1 - 2.0f * giou;
        }

        // Class cost via one-hot matmul: acc += (-prob) x onehot(id), K = 92.
        // B[k][n] layout mirrors A: lane = n + 16*(k>=2), vgpr = k&1.
        #pragma unroll
        for (int s = 0; s < KSLICES; ++s) {
            const int k0 = 4 * s + khalf;
            v2f b;
            b.x = (tid == k0    ) ? 1.0f : 0.0f;
            b.y = (tid == k0 + 1) ? 1.0f : 0.0f;
            acc = __builtin_amdgcn_wmma_f32_16x16x4_f32(
                /*neg_a=*/false, a[s], /*neg_b=*/false, b,
                /*c_mod=*/(short)0, acc, /*reuse_a=*/false, /*reuse_b=*/false);
        }

        // Store tile: lane owns (M = r + 8*half, N = nlane); NT stores (streaming 102MB)
        const size_t base = (size_t)(row_base + half * 8) * NT
                          + (size_t)(col_base0 + t * 16 + nlane);
        #pragma unroll
        for (int r = 0; r < 8; ++r)
            __builtin_nontemporal_store(acc[r], out + base + (size_t)r * NT);
    }
}

extern "C" void kernel_launch(void* const* d_in, const int* in_sizes, int n_in,
                              void* d_out, int out_size, void* d_ws, size_t ws_size,
                              hipStream_t stream) {
    (void)in_sizes; (void)n_in; (void)d_ws; (void)ws_size; (void)out_size;
    const float* logits = (const float*)d_in[0];   // (16,1000,92) f32
    const float* pboxes = (const float*)d_in[1];   // (16,1000,4)  f32
    const int*   tids   = (const int*)  d_in[2];   // (1600,)      i32
    const float* tboxes = (const float*)d_in[3];   // (1600,4)     f32
    float* out = (float*)d_out;                    // (16,1000,1600) f32

    const int nblocks = (NTASKS + WAVES_PER_BLOCK - 1) / WAVES_PER_BLOCK;  // 1250
    hipLaunchKernelGGL(matcher_cost_kernel, dim3(nblocks), dim3(WAVES_PER_BLOCK * 32),
                       0, stream, logits, pboxes, tids, tboxes, out);
}